// MIoU_41326175322262
// MI455X (gfx1250) — compile-verified
//
#include <hip/hip_runtime.h>
#include <hip/hip_bf16.h>

// mIoU over 21 classes, shape (32,512,512) int32 pred/gt -> scalar f32.
// Pass 1: memory-bound 3-way histogram (per-wave private LDS bins, ds_add_u32),
//         one f32 partial row of 96 bins per workgroup into d_ws.
// Pass 2: reduction of partial rows via V_WMMA_F32_16X16X4_F32 (ones x P),
//         exact in f32 since all counts < 2^24, then IoU mean on one lane.

#define NCLASS 21
#define PB     32            // padded bins per histogram
#define NBINS  96            // 3 histograms * 32
#define WAVES_PER_BLOCK 8    // 256 threads, wave32

typedef float v2f __attribute__((ext_vector_type(2)));
typedef float v8f __attribute__((ext_vector_type(8)));

__global__ __launch_bounds__(256) void miou_hist_kernel(
    const int* __restrict__ pred, const int* __restrict__ gt,
    float* __restrict__ partial, int n)
{
    __shared__ unsigned int sh[WAVES_PER_BLOCK * NBINS];
    const int tid  = threadIdx.x;
    const int wave = tid >> 5;

    for (int i = tid; i < WAVES_PER_BLOCK * NBINS; i += blockDim.x) sh[i] = 0u;
    __syncthreads();

    unsigned int* h = sh + wave * NBINS;

    const int gidx   = blockIdx.x * blockDim.x + tid;
    const int stride = gridDim.x * blockDim.x;
    const int n4     = n >> 2;
    const int4* __restrict__ p4 = (const int4*)pred;
    const int4* __restrict__ g4 = (const int4*)gt;

    // Vectorized main loop: global_load_b128 per input, 3 ds_add per element.
    for (int i = gidx; i < n4; i += stride) {
        const int4 p = p4[i];
        const int4 g = g4[i];
        atomicAdd(&h[p.x], 1u); atomicAdd(&h[PB + g.x], 1u); if (p.x == g.x) atomicAdd(&h[2*PB + p.x], 1u);
        atomicAdd(&h[p.y], 1u); atomicAdd(&h[PB + g.y], 1u); if (p.y == g.y) atomicAdd(&h[2*PB + p.y], 1u);
        atomicAdd(&h[p.z], 1u); atomicAdd(&h[PB + g.z], 1u); if (p.z == g.z) atomicAdd(&h[2*PB + p.z], 1u);
        atomicAdd(&h[p.w], 1u); atomicAdd(&h[PB + g.w], 1u); if (p.w == g.w) atomicAdd(&h[2*PB + p.w], 1u);
    }
    // Scalar tail (n not multiple of 4).
    for (int i = (n4 << 2) + gidx; i < n; i += stride) {
        const int p = pred[i];
        const int g = gt[i];
        atomicAdd(&h[p], 1u); atomicAdd(&h[PB + g], 1u); if (p == g) atomicAdd(&h[2*PB + p], 1u);
    }
    __syncthreads();

    // Collapse the 8 per-wave histograms into this block's f32 partial row.
    for (int i = tid; i < NBINS; i += blockDim.x) {
        unsigned int s = 0u;
        #pragma unroll
        for (int w = 0; w < WAVES_PER_BLOCK; ++w) s += sh[w * NBINS + i];
        partial[blockIdx.x * NBINS + i] = (float)s;   // exact: s < 2^24
    }
}

// 6 waves (192 threads): wave t reduces columns [16t, 16t+16) of the
// nrows x 96 partial matrix with chained V_WMMA_F32_16X16X4_F32 (A = ones).
__global__ __launch_bounds__(192) void miou_reduce_kernel(
    const float* __restrict__ partial, float* __restrict__ out, int nrows)
{
    __shared__ float sums[NBINS];

    const int tid  = threadIdx.x;
    const int wave = tid >> 5;            // 0..5 -> column tile
    const int lane = tid & 31;
    const int col  = wave * 16 + (lane & 15);
    const int rsel = (lane < 16) ? 0 : 2; // half-wave row-pair selector

    v2f a; a[0] = 1.0f; a[1] = 1.0f;      // 16x4 all-ones A matrix
    v8f c = {};                           // f32 accumulator (16x16 C/D)

    for (int chunk = 0; chunk < nrows; chunk += 4) {
        const int r0 = chunk + rsel;
        v2f b;
        b[0] = partial[(r0 + 0) * NBINS + col];
        b[1] = partial[(r0 + 1) * NBINS + col];
        // D = A(16x4 ones) * B(4x16 partial rows) + C  -> column sums
        c = __builtin_amdgcn_wmma_f32_16x16x4_f32(
                /*neg_a=*/false, a, /*neg_b=*/false, b,
                /*c_mod=*/(short)0, c, /*reuse_a=*/false, /*reuse_b=*/false);
    }

    // D row M=0 lives in VGPR0 of lanes 0..15 (N = lane).
    if (lane < 16) sums[wave * 16 + lane] = c[0];
    __syncthreads();

    if (tid == 0) {
        float acc = 0.0f;
        #pragma unroll
        for (int k = 0; k < NCLASS; ++k) {
            const float P = sums[k];
            const float G = sums[PB + k];
            const float I = sums[2 * PB + k];
            const float U = P + G - I;
            acc += (I + 1e-7f) / (U + 1e-7f);
        }
        out[0] = acc / (float)NCLASS;
    }
}

extern "C" void kernel_launch(void* const* d_in, const int* in_sizes, int n_in,
                              void* d_out, int out_size, void* d_ws, size_t ws_size,
                              hipStream_t stream)
{
    const int* pred = (const int*)d_in[0];
    const int* gt   = (const int*)d_in[1];
    const int  n    = in_sizes[0];

    // Partial rows: as many as scratch allows, target 1024, multiple of 4.
    int nrows = 1024;
    const int maxrows = (int)(ws_size / (NBINS * sizeof(float))) & ~3;
    if (nrows > maxrows) nrows = maxrows;
    if (nrows < 4) nrows = 4;

    float* partial = (float*)d_ws;

    miou_hist_kernel  <<<nrows, 256, 0, stream>>>(pred, gt, partial, n);
    miou_reduce_kernel<<<1,     192, 0, stream>>>(partial, (float*)d_out, nrows);
}